// GNN_40029095198942
// MI455X (gfx1250) — compile-verified
//
#include <hip/hip_runtime.h>
#include <math.h>

#define NODES   200000
#define DEG     16
#define GRAPHS  4000
#define NPG     50
#define FIN     25
#define H1DIM   16
#define H2DIM   8

typedef __attribute__((ext_vector_type(2))) float v2f;
typedef __attribute__((ext_vector_type(8))) float v8f;

// ---------------------------------------------------------------------------
// K0: fold (w_lin @ w_post) and the degenerate degree-scaler triple (ratio==1
// since in-degree is structurally 16 for every node), fold biases, and pack
// the resulting weight matrices into the V_WMMA_F32_16X16X4_F32 B-operand
// lane layout:  lane<16 -> N=lane, regs {K0,K1}; lane>=16 -> N=lane-16, {K2,K3}.
// ---------------------------------------------------------------------------
__global__ __launch_bounds__(256) void fold_kernel(
    const float* __restrict__ w1_post, const float* __restrict__ b1_post,
    const float* __restrict__ w1_lin,  const float* __restrict__ b1_lin,
    const float* __restrict__ w2_post, const float* __restrict__ b2_post,
    const float* __restrict__ w2_lin,  const float* __restrict__ b2_lin,
    float* __restrict__ Bpack1, float* __restrict__ Bpack2,
    float* __restrict__ b1eff,  float* __restrict__ b2eff)
{
  int t = threadIdx.x;
  // Layer 1: W1eff[o][j] = sum_i w1_lin[o][i]*(w1_post[i][j]+w1_post[i][100+j]+w1_post[i][200+j])
  for (int idx = t; idx < 16 * 100; idx += 256) {
    int o = idx / 100, j = idx % 100;
    float acc = 0.f;
    for (int i = 0; i < 16; ++i) {
      float w = w1_lin[o * 16 + i];
      acc += w * (w1_post[i * 300 + j] + w1_post[i * 300 + 100 + j] + w1_post[i * 300 + 200 + j]);
    }
    int c = j >> 2, h = (j >> 1) & 1, u = j & 1;
    Bpack1[(c * 32 + (o + 16 * h)) * 2 + u] = acc;   // B[k=j][n=o]
  }
  // Layer 2: W2eff[o][j], o<8, j<64
  for (int idx = t; idx < 8 * 64; idx += 256) {
    int o = idx / 64, j = idx % 64;
    float acc = 0.f;
    for (int i = 0; i < 8; ++i) {
      float w = w2_lin[o * 8 + i];
      acc += w * (w2_post[i * 192 + j] + w2_post[i * 192 + 64 + j] + w2_post[i * 192 + 128 + j]);
    }
    int c = j >> 2, h = (j >> 1) & 1, u = j & 1;
    Bpack2[(c * 32 + (o + 16 * h)) * 2 + u] = acc;
  }
  // Zero the unused output columns (n = 8..15) of layer2's B tile.
  for (int idx = t; idx < 16 * 32 * 2; idx += 256) {
    int lane = (idx >> 1) & 31;
    if ((lane & 15) >= 8) Bpack2[idx] = 0.f;
  }
  if (t < 16) {
    float acc = b1_lin[t];
    for (int i = 0; i < 16; ++i) acc += w1_lin[t * 16 + i] * b1_post[i];
    b1eff[t] = acc;
  }
  if (t < 8) {
    float acc = b2_lin[t];
    for (int i = 0; i < 8; ++i) acc += w2_lin[t * 8 + i] * b2_post[i];
    b2eff[t] = acc;
  }
}

// ---------------------------------------------------------------------------
// K1: per-node pre-projection, layer 1.
// yd[n][f] = sum_i x[n][i]*w1_pre[f][i]      + b1_pre[f]   (dst half, bias folded)
// ys[n][f] = sum_i x[n][i]*w1_pre[f][25+i]                 (src half)
// ---------------------------------------------------------------------------
__global__ __launch_bounds__(256) void pre1_kernel(
    const float* __restrict__ x, const float* __restrict__ w1_pre,
    const float* __restrict__ b1_pre, float* __restrict__ yd, float* __restrict__ ys)
{
  int t = blockIdx.x * 256 + threadIdx.x;
  if (t >= NODES * 50) return;
  int n = t / 50, fo = t % 50;
  const float* xr = x + n * FIN;
  if (fo < FIN) {
    const float* wr = w1_pre + fo * 50;
    float acc = b1_pre[fo];
    for (int i = 0; i < FIN; ++i) acc += xr[i] * wr[i];
    yd[n * FIN + fo] = acc;
  } else {
    int f = fo - FIN;
    const float* wr = w1_pre + f * 50 + FIN;
    float acc = 0.f;
    for (int i = 0; i < FIN; ++i) acc += xr[i] * wr[i];
    ys[n * FIN + f] = acc;
  }
}

// ---------------------------------------------------------------------------
// K2: fused layer-1 aggregation (mean|min|max|std over 16 contiguous in-edges)
// + f32 WMMA post-GEMM [16 x 100] x [100 x 16] + bias + ReLU.
// One wave per 16-node tile; per-wave LDS staging (LDS in-order per wave).
// ---------------------------------------------------------------------------
__global__ __launch_bounds__(256) void layer1_kernel(
    const float* __restrict__ yd, const float* __restrict__ ys,
    const int* __restrict__ src, const float* __restrict__ Bpack1,
    const float* __restrict__ b1eff, float* __restrict__ h1)
{
  __shared__ float Atile[8][16][104];
  int wave = threadIdx.x >> 5;
  int lane = threadIdx.x & 31;
  int tile = blockIdx.x * 8 + wave;
  if (tile >= NODES / 16) return;
  int base = tile * 16;
  float (&A)[16][104] = Atile[wave];

  int f  = lane;                        // feature owned by this lane (valid if <25)
  int fc = (f < FIN) ? f : 0;           // clamp for safe addressing
  for (int i = 0; i < 16; ++i) {
    int n = base + i;
    float a0 = yd[n * FIN + fc];
    float s = 0.f, s2 = 0.f, mn = 3.0e38f, mx = -3.0e38f;
    for (int k = 0; k < DEG; ++k) {
      int sj = __builtin_amdgcn_readfirstlane(src[n * DEG + k]);  // wave-uniform -> SGPR
      float v = a0 + ys[sj * FIN + fc];
      s += v; s2 += v * v;
      mn = fminf(mn, v); mx = fmaxf(mx, v);
    }
    if (f < FIN) {
      float mean  = s  * 0.0625f;
      float mean2 = s2 * 0.0625f;
      float var   = mean2 - mean * mean;
      var = var > 0.f ? var : 0.f;
      A[i][f]      = mean;
      A[i][25 + f] = mn;
      A[i][50 + f] = mx;
      A[i][75 + f] = sqrtf(var + 1e-5f);
    }
  }
  __builtin_amdgcn_wave_barrier();

  int m = lane & 15, h = lane >> 4;     // A layout: lane<16 -> {K0,K1}, lane>=16 -> {K2,K3}
  v8f c;
  float bn = b1eff[m];                  // bias per output column n = m
  for (int r = 0; r < 8; ++r) c[r] = bn;
  for (int ck = 0; ck < 25; ++ck) {     // K = 100 = 25 x 4, exact fit, full f32
    v2f a, b;
    a.x = A[m][4 * ck + 2 * h];
    a.y = A[m][4 * ck + 2 * h + 1];
    b = *(const v2f*)(Bpack1 + (ck * 32 + lane) * 2);
    c = __builtin_amdgcn_wmma_f32_16x16x4_f32(false, a, false, b, (short)0, c, false, false);
  }
  for (int r = 0; r < 8; ++r) {         // D: lane holds column n=m, rows r+8*h
    float v = c[r]; v = v > 0.f ? v : 0.f;
    h1[(base + r + 8 * h) * H1DIM + m] = v;
  }
}

// ---------------------------------------------------------------------------
// K3: per-node pre-projection, layer 2 (h1 [N,16] -> yd2/ys2 [N,16]).
// ---------------------------------------------------------------------------
__global__ __launch_bounds__(256) void pre2_kernel(
    const float* __restrict__ h1, const float* __restrict__ w2_pre,
    const float* __restrict__ b2_pre, float* __restrict__ yd2, float* __restrict__ ys2)
{
  int t = blockIdx.x * 256 + threadIdx.x;
  if (t >= NODES * 32) return;
  int n = t >> 5, fo = t & 31;
  const float* hr = h1 + n * H1DIM;
  if (fo < 16) {
    const float* wr = w2_pre + fo * 32;
    float acc = b2_pre[fo];
    for (int i = 0; i < 16; ++i) acc += hr[i] * wr[i];
    yd2[n * 16 + fo] = acc;
  } else {
    int fb = fo - 16;
    const float* wr = w2_pre + fb * 32 + 16;
    float acc = 0.f;
    for (int i = 0; i < 16; ++i) acc += hr[i] * wr[i];
    ys2[n * 16 + fb] = acc;
  }
}

// ---------------------------------------------------------------------------
// K4: fused layer-2 aggregation + f32 WMMA post-GEMM [16x64]x[64x16(8 live)].
// ---------------------------------------------------------------------------
__global__ __launch_bounds__(256) void layer2_kernel(
    const float* __restrict__ yd2, const float* __restrict__ ys2,
    const int* __restrict__ src, const float* __restrict__ Bpack2,
    const float* __restrict__ b2eff, float* __restrict__ h2)
{
  __shared__ float Atile[8][16][68];
  int wave = threadIdx.x >> 5;
  int lane = threadIdx.x & 31;
  int tile = blockIdx.x * 8 + wave;
  if (tile >= NODES / 16) return;
  int base = tile * 16;
  float (&A)[16][68] = Atile[wave];

  int f = lane & 15, h = lane >> 4;     // two nodes per iteration, 16 features each
  for (int i = 0; i < 16; i += 2) {
    int n = base + i + h;
    float a0 = yd2[n * 16 + f];
    float s = 0.f, s2 = 0.f, mn = 3.0e38f, mx = -3.0e38f;
    for (int k = 0; k < DEG; ++k) {
      int sj = src[n * DEG + k];
      float v = a0 + ys2[sj * 16 + f];
      s += v; s2 += v * v;
      mn = fminf(mn, v); mx = fmaxf(mx, v);
    }
    float mean  = s  * 0.0625f;
    float mean2 = s2 * 0.0625f;
    float var   = mean2 - mean * mean;
    var = var > 0.f ? var : 0.f;
    A[i + h][f]      = mean;
    A[i + h][16 + f] = mn;
    A[i + h][32 + f] = mx;
    A[i + h][48 + f] = sqrtf(var + 1e-5f);
  }
  __builtin_amdgcn_wave_barrier();

  int m = f;
  v8f c;
  float bn = (m < 8) ? b2eff[m] : 0.f;
  for (int r = 0; r < 8; ++r) c[r] = bn;
  for (int ck = 0; ck < 16; ++ck) {     // K = 64 = 16 x 4
    v2f a, b;
    a.x = A[m][4 * ck + 2 * h];
    a.y = A[m][4 * ck + 2 * h + 1];
    b = *(const v2f*)(Bpack2 + (ck * 32 + lane) * 2);
    c = __builtin_amdgcn_wmma_f32_16x16x4_f32(false, a, false, b, (short)0, c, false, false);
  }
  if (m < 8) {
    for (int r = 0; r < 8; ++r) {
      float v = c[r]; v = v > 0.f ? v : 0.f;
      h2[(base + r + 8 * h) * H2DIM + m] = v;
    }
  }
}

// ---------------------------------------------------------------------------
// K5: global_add_pool over 50 contiguous nodes + FC [8->2] + log_softmax.
// ---------------------------------------------------------------------------
__global__ __launch_bounds__(256) void pool_kernel(
    const float* __restrict__ h2, const float* __restrict__ fc_w,
    const float* __restrict__ fc_b, float* __restrict__ out)
{
  int g = blockIdx.x * 256 + threadIdx.x;
  if (g >= GRAPHS) return;
  float p[8];
  for (int o = 0; o < 8; ++o) p[o] = 0.f;
  const float* hr = h2 + (size_t)g * NPG * H2DIM;
  for (int n = 0; n < NPG; ++n)
    for (int o = 0; o < 8; ++o) p[o] += hr[n * 8 + o];
  float l0 = fc_b[0], l1 = fc_b[1];
  for (int o = 0; o < 8; ++o) { l0 += p[o] * fc_w[o]; l1 += p[o] * fc_w[8 + o]; }
  float mx = fmaxf(l0, l1);
  float lse = mx + logf(expf(l0 - mx) + expf(l1 - mx));
  out[g * 2 + 0] = l0 - lse;
  out[g * 2 + 1] = l1 - lse;
}

// ---------------------------------------------------------------------------
extern "C" void kernel_launch(void* const* d_in, const int* in_sizes, int n_in,
                              void* d_out, int out_size, void* d_ws, size_t ws_size,
                              hipStream_t stream)
{
  (void)in_sizes; (void)n_in; (void)out_size; (void)ws_size;
  const float* x       = (const float*)d_in[0];
  const int*   src     = (const int*)d_in[1];      // edge_index[0] = src (first E ints)
  const float* w1_pre  = (const float*)d_in[3];
  const float* b1_pre  = (const float*)d_in[4];
  const float* w1_post = (const float*)d_in[5];
  const float* b1_post = (const float*)d_in[6];
  const float* w1_lin  = (const float*)d_in[7];
  const float* b1_lin  = (const float*)d_in[8];
  const float* w2_pre  = (const float*)d_in[9];
  const float* b2_pre  = (const float*)d_in[10];
  const float* w2_post = (const float*)d_in[11];
  const float* b2_post = (const float*)d_in[12];
  const float* w2_lin  = (const float*)d_in[13];
  const float* b2_lin  = (const float*)d_in[14];
  const float* fc_w    = (const float*)d_in[15];
  const float* fc_b    = (const float*)d_in[16];
  float* out = (float*)d_out;

  float* ws     = (float*)d_ws;
  float* Bpack1 = ws;                         // 25*32*2 = 1600 floats
  float* Bpack2 = ws + 1600;                  // 16*32*2 = 1024
  float* b1eff  = ws + 2624;                  // 16
  float* b2eff  = ws + 2640;                  // 8
  float* yd1    = ws + 4096;                  // N*25
  float* ys1    = yd1 + (size_t)NODES * FIN;  // N*25
  float* h1     = ys1 + (size_t)NODES * FIN;  // N*16
  float* h2     = h1  + (size_t)NODES * H1DIM;// N*8
  float* yd2    = yd1;  // alias: yd1/ys1 dead after layer1_kernel
  float* ys2    = ys1;

  fold_kernel<<<1, 256, 0, stream>>>(w1_post, b1_post, w1_lin, b1_lin,
                                     w2_post, b2_post, w2_lin, b2_lin,
                                     Bpack1, Bpack2, b1eff, b2eff);
  pre1_kernel<<<(NODES * 50 + 255) / 256, 256, 0, stream>>>(x, w1_pre, b1_pre, yd1, ys1);
  layer1_kernel<<<(NODES / 16 + 7) / 8, 256, 0, stream>>>(yd1, ys1, src, Bpack1, b1eff, h1);
  pre2_kernel<<<(NODES * 32 + 255) / 256, 256, 0, stream>>>(h1, w2_pre, b2_pre, yd2, ys2);
  layer2_kernel<<<(NODES / 16 + 7) / 8, 256, 0, stream>>>(yd2, ys2, src, Bpack2, b2eff, h2);
  pool_kernel<<<(GRAPHS + 255) / 256, 256, 0, stream>>>(h2, fc_w, fc_b, out);
}